// lossFunc_14731737825626
// MI455X (gfx1250) — compile-verified
//
#include <hip/hip_runtime.h>
#include <math.h>
#include <stdint.h>

// Problem geometry (from reference): pred (256,199,1000), batch (256,200,1000)
#define NQ      1000
#define NT4     250      // float4 chunks per row (1000/4) for async staging
#define TSTEPS  199
#define NBATCH  256

// ---------------------------------------------------------------------------
// Kernel A: one workgroup per (student b, step t) row.
//  - gt row (batch[b, t+1, :], 4000 B, 16B-aligned) staged to LDS with the
//    CDNA5 async-to-LDS path (ASYNCcnt, b128 per lane), overlapped with
//    coalesced NT b32 loads of pred.
//  - Element mapping e = tid + 256k so every b32 load/store instruction is a
//    single contiguous 128 B line per wave (full-line writes for the 408 MB
//    of streamed output; output base d_out+1 is only 4 B aligned so b32 is
//    the widest legal store).
//  - row mask via LDS atomicOr; masked BCE sum block-reduced into rowsum[row].
// ---------------------------------------------------------------------------
__global__ __launch_bounds__(256) void bce_row_kernel(
    const float* __restrict__ pred,
    const float* __restrict__ batch,
    float* __restrict__ out_pred,   // d_out + 1                 (256*199*1000)
    float* __restrict__ out_gt,     // d_out + 1 + R*NQ          (256*199*1000)
    float* __restrict__ out_mask,   // d_out + 1 + 2*R*NQ        (256*199)
    float* __restrict__ rowsum)     // ws                         (256*199)
{
    const int t   = blockIdx.x;     // 0..198
    const int b   = blockIdx.y;     // 0..255
    const int tid = threadIdx.x;    // 0..255
    const int row = b * TSTEPS + t;

    __shared__ __align__(16) float s_gt[NQ];
    __shared__ int   s_flag;
    __shared__ float s_wsum[8];

    const float* gt_row = batch + ((size_t)b * 200 + (t + 1)) * NQ; // 16B aligned
    const float* pr_row = pred  + (size_t)row * NQ;                 // 16B aligned

    // 1) Async-stage the gt row into LDS (global -> LDS, tracked by ASYNCcnt).
    //    250 lanes x 16 B; LDS dest = low 32 bits of the flat address of the
    //    __shared__ slot (= group-relative LDS byte offset).
    if (tid < NT4) {
        unsigned lds_addr = (unsigned)(uintptr_t)(&s_gt[tid * 4]);
        const float* gsrc = gt_row + tid * 4;
        asm volatile("global_load_async_to_lds_b128 %0, %1, off"
                     :: "v"(lds_addr), "v"(gsrc)
                     : "memory");
    }

    // 2) Concurrently pull the pred row into VGPRs: 4 coalesced NT b32 loads
    //    (pure stream, keep it out of L2's way).
    float p[4] = {0.f, 0.f, 0.f, 0.f};
    #pragma unroll
    for (int k = 0; k < 4; ++k) {
        const int e = tid + 256 * k;
        if (e < NQ) p[k] = __builtin_nontemporal_load(pr_row + e);
    }

    if (tid == 0) s_flag = 0;

    // Wait for this wave's async copies, then workgroup barrier => whole row
    // + cleared flag visible in LDS.
    asm volatile("s_wait_asynccnt 0x0" ::: "memory");
    __syncthreads();

    // 3) Conflict-free stride-1 LDS readback + mask predicate.
    float g[4] = {0.f, 0.f, 0.f, 0.f};
    bool has1 = false;
    #pragma unroll
    for (int k = 0; k < 4; ++k) {
        const int e = tid + 256 * k;
        if (e < NQ) {
            g[k] = s_gt[e];
            has1 |= (g[k] == 1.0f);
        }
    }
    if (has1) atomicOr(&s_flag, 1);   // expected ~2 hot lanes per row
    __syncthreads();

    const float m = s_flag ? 1.0f : 0.0f;

    // 4) Masked outputs: fully coalesced NT b32 stores (full 128 B lines).
    const size_t rbase = (size_t)row * NQ;
    #pragma unroll
    for (int k = 0; k < 4; ++k) {
        const int e = tid + 256 * k;
        if (e < NQ) {
            __builtin_nontemporal_store(p[k] * m, out_pred + rbase + e);
            __builtin_nontemporal_store(g[k] * m, out_gt   + rbase + e);
        }
    }

    // 5) Masked BCE partial sum (gt is exactly 0.0 or 1.0).
    float local = 0.0f;
    if (s_flag) {
        #pragma unroll
        for (int k = 0; k < 4; ++k) {
            const int e = tid + 256 * k;
            if (e < NQ) {
                const float lp = fmaxf(logf(p[k]),    -100.0f);
                const float l1 = fmaxf(log1pf(-p[k]), -100.0f);
                local += (g[k] == 1.0f) ? -lp : -l1;
            }
        }
    }

    // 6) Deterministic block reduction.
    #pragma unroll
    for (int off = 16; off > 0; off >>= 1)
        local += __shfl_down(local, off, 32);
    if ((tid & 31) == 0) s_wsum[tid >> 5] = local;
    __syncthreads();
    if (tid == 0) {
        float tot = 0.0f;
        #pragma unroll
        for (int w = 0; w < 8; ++w) tot += s_wsum[w];
        rowsum[row]   = tot;   // 0 when row not masked
        out_mask[row] = m;     // row_mask as 0.0/1.0
    }
}

// ---------------------------------------------------------------------------
// Kernel B: per-student normalization. Block b reduces its 199 rows.
// ---------------------------------------------------------------------------
__global__ __launch_bounds__(256) void per_student_kernel(
    const float* __restrict__ rowsum,
    const float* __restrict__ mask,
    float* __restrict__ perstudent)
{
    const int b   = blockIdx.x;
    const int tid = threadIdx.x;
    float s = 0.0f, c = 0.0f;
    if (tid < TSTEPS) {
        s = rowsum[b * TSTEPS + tid];
        c = mask  [b * TSTEPS + tid];
    }
    #pragma unroll
    for (int off = 16; off > 0; off >>= 1) {
        s += __shfl_down(s, off, 32);
        c += __shfl_down(c, off, 32);
    }
    __shared__ float ss[8], sc[8];
    if ((tid & 31) == 0) { ss[tid >> 5] = s; sc[tid >> 5] = c; }
    __syncthreads();
    if (tid == 0) {
        float S = 0.0f, C = 0.0f;
        #pragma unroll
        for (int w = 0; w < 8; ++w) { S += ss[w]; C += sc[w]; }
        // n_elem = (#masked rows) * 1000 ; C >= 1 since batch[:,1,0] == 1
        perstudent[b] = S / (C * (float)NQ);
    }
}

// ---------------------------------------------------------------------------
// Kernel C: final sum of 256 per-student losses -> d_out[0].
// ---------------------------------------------------------------------------
__global__ __launch_bounds__(256) void loss_kernel(
    const float* __restrict__ perstudent,
    float* __restrict__ out_loss)
{
    const int tid = threadIdx.x;
    float v = perstudent[tid];
    #pragma unroll
    for (int off = 16; off > 0; off >>= 1)
        v += __shfl_down(v, off, 32);
    __shared__ float ss[8];
    if ((tid & 31) == 0) ss[tid >> 5] = v;
    __syncthreads();
    if (tid == 0) {
        float S = 0.0f;
        #pragma unroll
        for (int w = 0; w < 8; ++w) S += ss[w];
        out_loss[0] = S;
    }
}

// ---------------------------------------------------------------------------
extern "C" void kernel_launch(void* const* d_in, const int* in_sizes, int n_in,
                              void* d_out, int out_size, void* d_ws, size_t ws_size,
                              hipStream_t stream) {
    const float* pred  = (const float*)d_in[0];   // (256,199,1000) f32
    const float* batch = (const float*)d_in[1];   // (256,200,1000) f32
    float* out = (float*)d_out;

    const size_t R = (size_t)NBATCH * TSTEPS;     // 50944 rows
    float* out_pred = out + 1;
    float* out_gt   = out + 1 + R * NQ;
    float* out_mask = out + 1 + 2 * R * NQ;

    float* rowsum     = (float*)d_ws;             // R floats
    float* perstudent = rowsum + R;               // 256 floats

    dim3 gridA(TSTEPS, NBATCH);
    bce_row_kernel<<<gridA, 256, 0, stream>>>(pred, batch, out_pred, out_gt,
                                              out_mask, rowsum);
    per_student_kernel<<<NBATCH, 256, 0, stream>>>(rowsum, out_mask, perstudent);
    loss_kernel<<<1, 256, 0, stream>>>(perstudent, out);
}